// GCNLayer_74491912781904
// MI455X (gfx1250) — compile-verified
//
#include <hip/hip_runtime.h>
#include <hip/hip_bf16.h>

typedef __attribute__((ext_vector_type(2))) float v2f;
typedef __attribute__((ext_vector_type(8))) float v8f;

// ---------------------------------------------------------------------------
// Kernel 1: zero agg accumulator (d_out) and degree buffer (d_ws)
// ---------------------------------------------------------------------------
__global__ void gcn_zero_kernel(float* __restrict__ agg, float* __restrict__ deg,
                                long long n_agg, int n_deg) {
    long long i = (long long)blockIdx.x * blockDim.x + threadIdx.x;
    if (i < n_agg) agg[i] = 0.0f;
    if (i < n_deg) deg[i] = 0.0f;
}

// ---------------------------------------------------------------------------
// Kernel 2: degree of each destination node (scatter-add of ones over row)
// ---------------------------------------------------------------------------
__global__ void gcn_degree_kernel(const int* __restrict__ edge_index,
                                  float* __restrict__ deg, int E) {
    int e = blockIdx.x * blockDim.x + threadIdx.x;
    if (e < E) {
        atomicAdd(&deg[edge_index[e]], 1.0f);  // edge_index[0][e] = row
    }
}

// ---------------------------------------------------------------------------
// Kernel 3: deg -> deg^-0.5 in place (0 where deg == 0)
// ---------------------------------------------------------------------------
__global__ void gcn_dinv_kernel(float* __restrict__ deg, int N) {
    int i = blockIdx.x * blockDim.x + threadIdx.x;
    if (i < N) {
        float d = deg[i];
        deg[i] = (d > 0.0f) ? rsqrtf(d) : 0.0f;
    }
}

// ---------------------------------------------------------------------------
// Kernel 4: edge scatter. 32 lanes per edge; lane L handles features 2L,2L+1.
// Gather of x row is a fully coalesced 256B line; atomics land on one line.
// agg[row] += x[col] * dinv[col]
// ---------------------------------------------------------------------------
__global__ void gcn_scatter_kernel(const float* __restrict__ x,
                                   const int* __restrict__ edge_index,
                                   const float* __restrict__ dinv,
                                   float* __restrict__ agg, int E) {
    long long gid = (long long)blockIdx.x * blockDim.x + threadIdx.x;
    long long e = gid >> 5;
    int lane = (int)(gid & 31);
    if (e >= E) return;

    int dst = edge_index[e];              // row
    int src = edge_index[(long long)E + e];  // col
    float s = dinv[src];

    float2 v = *(const float2*)(x + (size_t)src * 64 + 2 * lane);
    float* outp = agg + (size_t)dst * 64 + 2 * lane;
    atomicAdd(outp + 0, v.x * s);
    atomicAdd(outp + 1, v.y * s);
}

// ---------------------------------------------------------------------------
// Kernel 5: out = relu( (agg * dinv[row]) @ W^T + b ) via V_WMMA_F32_16X16X4_F32
// One wave computes a 16x64 output tile (4 accumulators of 16x16).
// W (64x64 f32, 16KB) staged in LDS per 4-wave block. In-place over agg:
// each wave reads its 16 rows into registers before any store.
//
// A fragment (16x4 f32, 2 VGPRs): lanes 0-15 hold row M=lane, K = 4k,4k+1;
//                                 lanes 16-31 hold row M=lane-16, K = 4k+2,4k+3.
// B fragment (4x16 f32, 2 VGPRs): lanes 0-15 hold col N, K = 4k,4k+1; etc.
// B(K,N) = W[N*64 + K]  (since out = agg @ W^T).
// C/D (16x16 f32, 8 VGPRs): VGPR i -> M=i (lanes 0-15) / M=i+8 (lanes 16-31).
// ---------------------------------------------------------------------------
__global__ __launch_bounds__(128) void gcn_wmma_gemm_kernel(
    const float* __restrict__ dinv, const float* __restrict__ W,
    const float* __restrict__ bias, float* __restrict__ out,
    int N, int nTiles) {
    __shared__ float Wlds[64 * 64];
    __shared__ float blds[64];

    int tid = threadIdx.x;
    // Cooperative load of W into LDS (4096 floats / 128 threads, float4)
    {
        const float4* W4 = (const float4*)W;
        float4* Wl4 = (float4*)Wlds;
        #pragma unroll
        for (int i = 0; i < 8; ++i) Wl4[tid + i * 128] = W4[tid + i * 128];
        if (tid < 64) blds[tid] = bias[tid];
    }
    __syncthreads();

    int wave = tid >> 5;
    int lane = tid & 31;
    int tile = blockIdx.x * 4 + wave;
    if (tile < nTiles) {                 // wave-uniform: EXEC all-ones for WMMA
        int r0 = tile * 16;
        int half = lane >> 4;            // 0: K%4 in {0,1}; 1: K%4 in {2,3}
        int ln = lane & 15;
        int row = r0 + ln;
        bool rowOK = row < N;
        float s = rowOK ? dinv[row] : 0.0f;

        // Load + scale A fragments (agg lives in `out`)
        v2f a[16];
        const float* Arow = out + (size_t)row * 64 + 2 * half;
        #pragma unroll
        for (int k = 0; k < 16; ++k) {
            float2 v = rowOK ? *(const float2*)(Arow + 4 * k)
                             : make_float2(0.0f, 0.0f);
            a[k] = (v2f){v.x * s, v.y * s};
        }

        v8f acc[4] = {v8f{}, v8f{}, v8f{}, v8f{}};
        #pragma unroll
        for (int t = 0; t < 4; ++t) {
            #pragma unroll
            for (int k = 0; k < 16; ++k) {
                float2 bv = *(const float2*)&Wlds[(16 * t + ln) * 64 + 4 * k + 2 * half];
                v2f bfrag = (v2f){bv.x, bv.y};
                acc[t] = __builtin_amdgcn_wmma_f32_16x16x4_f32(
                    /*neg_a=*/false, a[k], /*neg_b=*/false, bfrag,
                    /*c_mod=*/(short)0, acc[t],
                    /*reuse_a=*/false, /*reuse_b=*/false);
            }
        }

        // Epilogue: bias + relu, store (overwrites the same 16 rows)
        #pragma unroll
        for (int t = 0; t < 4; ++t) {
            float bcol = blds[16 * t + ln];
            #pragma unroll
            for (int i = 0; i < 8; ++i) {
                int rl = half ? (i + 8) : i;
                int r = r0 + rl;
                if (r < N) {
                    float v = acc[t][i] + bcol;
                    out[(size_t)r * 64 + 16 * t + ln] = fmaxf(v, 0.0f);
                }
            }
        }
    }
}

// ---------------------------------------------------------------------------
extern "C" void kernel_launch(void* const* d_in, const int* in_sizes, int n_in,
                              void* d_out, int out_size, void* d_ws, size_t ws_size,
                              hipStream_t stream) {
    const float* x  = (const float*)d_in[0];        // [N, 64] f32
    const int*   ei = (const int*)d_in[1];          // [2, E]  int
    const float* W  = (const float*)d_in[2];        // [64, 64] f32
    const float* b  = (const float*)d_in[3];        // [64] f32
    float* out = (float*)d_out;                     // [N, 64] f32 (also agg)
    float* deg = (float*)d_ws;                      // [N] f32 scratch

    int N = in_sizes[0] / 64;
    int E = in_sizes[1] / 2;

    long long n_agg = (long long)N * 64;
    {
        long long threads = n_agg;
        int blocks = (int)((threads + 255) / 256);
        gcn_zero_kernel<<<blocks, 256, 0, stream>>>(out, deg, n_agg, N);
    }
    gcn_degree_kernel<<<(E + 255) / 256, 256, 0, stream>>>(ei, deg, E);
    gcn_dinv_kernel<<<(N + 255) / 256, 256, 0, stream>>>(deg, N);
    {
        long long threads = (long long)E * 32;
        int blocks = (int)((threads + 255) / 256);
        gcn_scatter_kernel<<<blocks, 256, 0, stream>>>(x, ei, deg, out, E);
    }
    {
        int nTiles = (N + 15) / 16;
        int blocks = (nTiles + 3) / 4;
        gcn_wmma_gemm_kernel<<<blocks, 128, 0, stream>>>(deg, W, b, out, N, nTiles);
    }
}